// Network_88914412961962
// MI455X (gfx1250) — compile-verified
//
#include <hip/hip_runtime.h>
#include <hip/hip_bf16.h>

typedef __attribute__((ext_vector_type(2))) float v2f;
typedef __attribute__((ext_vector_type(8))) float v8f;

#define NPTS   8192
#define NPOINT 2048
#define BATCH  4
#define RAD2   0.04f

// ---- CDNA5 async global->LDS path (GLOBAL_LOAD_ASYNC_TO_LDS_B32, ASYNCcnt) ----
#if defined(__has_builtin)
#if __has_builtin(__builtin_amdgcn_global_load_async_to_lds_b32)
#define HAVE_ASYNC_LDS 1
#endif
#endif
#ifndef HAVE_ASYNC_LDS
#define HAVE_ASYNC_LDS 0
#endif

typedef __attribute__((address_space(1))) int as1_int;
typedef __attribute__((address_space(3))) int as3_int;

__device__ __forceinline__ void async_g2l_b32(const float* g, float* l)
{
#if HAVE_ASYNC_LDS
    __builtin_amdgcn_global_load_async_to_lds_b32(
        (as1_int*)(float*)g,           // global src (addrspace(1) int*)
        (as3_int*)l,                   // LDS dst (addrspace(3) int*)
        0, 0);
#else
    *l = *g;                           // fallback: VGPR bounce
#endif
}

__device__ __forceinline__ void async_wait0()
{
#if HAVE_ASYNC_LDS
#if __has_builtin(__builtin_amdgcn_s_wait_asynccnt)
    __builtin_amdgcn_s_wait_asynccnt(0);
#else
    asm volatile("s_wait_asynccnt 0x0" ::: "memory");
#endif
#endif
}

// ---------------------------------------------------------------------------
// Kernel 1: farthest point sampling. One 1024-thread block per batch.
// All 8192 points live in LDS (96KB of the 320KB WGP LDS); per-thread running
// distances live in registers (8 each). Emits new_xyz[b][it] = xyz[farthest].
// ---------------------------------------------------------------------------
__global__ void __launch_bounds__(1024)
fps_kernel(const float* __restrict__ xyz, float* __restrict__ new_xyz)
{
    __shared__ float P[3 * NPTS];      // 96 KB
    __shared__ float redv[32];
    __shared__ int   redi[32];
    __shared__ int   sf;

    const int b = blockIdx.x, tid = threadIdx.x;
    for (int e = tid; e < 3 * NPTS; e += 1024) P[e] = xyz[b * 3 * NPTS + e];
    if (tid == 0) sf = 0;

    float d[8];
#pragma unroll
    for (int i = 0; i < 8; ++i) d[i] = 1e10f;
    __syncthreads();

    const int lane = tid & 31, wv = tid >> 5;
    for (int it = 0; it < NPOINT; ++it) {
        const int f = sf;                               // uniform
        const float cx = P[3 * f], cy = P[3 * f + 1], cz = P[3 * f + 2];
        if (tid == 0) {
            new_xyz[(b * NPOINT + it) * 3 + 0] = cx;
            new_xyz[(b * NPOINT + it) * 3 + 1] = cy;
            new_xyz[(b * NPOINT + it) * 3 + 2] = cz;
        }
        float bv = -1.0f; int bi = 0;
#pragma unroll
        for (int i = 0; i < 8; ++i) {
            const int n = tid + i * 1024;
            const float dx = P[3 * n] - cx, dy = P[3 * n + 1] - cy, dz = P[3 * n + 2] - cz;
            const float dd = dx * dx + dy * dy + dz * dz;
            const float nd = d[i] < dd ? d[i] : dd;
            d[i] = nd;
            if (nd > bv) { bv = nd; bi = n; }           // strict > : lowest index wins ties
        }
        // wave32 argmax (prefer lower index on ties, matching jnp.argmax)
        for (int off = 16; off; off >>= 1) {
            const float ov = __shfl_down(bv, off, 32);
            const int   oi = __shfl_down(bi, off, 32);
            if (ov > bv || (ov == bv && oi < bi)) { bv = ov; bi = oi; }
        }
        if (lane == 0) { redv[wv] = bv; redi[wv] = bi; }
        __syncthreads();
        if (tid == 0) {
            float mv = redv[0]; int mi = redi[0];
            for (int w = 1; w < 32; ++w)
                if (redv[w] > mv || (redv[w] == mv && redi[w] < mi)) { mv = redv[w]; mi = redi[w]; }
            sf = mi;
        }
        __syncthreads();
    }
}

// ---------------------------------------------------------------------------
// Kernel 2: ball query. One wave per query point; ordered first-16 extraction
// via ballot + popc prefix, early exit once 16 neighbors found.
// ---------------------------------------------------------------------------
__global__ void __launch_bounds__(256)
ballquery_kernel(const float* __restrict__ xyz, const float* __restrict__ new_xyz,
                 int* __restrict__ nidx)
{
    const int wid  = (blockIdx.x * blockDim.x + threadIdx.x) >> 5;   // query id
    const int lane = threadIdx.x & 31;
    const int b = wid >> 11;
    const float qx = new_xyz[wid * 3 + 0];
    const float qy = new_xyz[wid * 3 + 1];
    const float qz = new_xyz[wid * 3 + 2];
    int* row = &nidx[wid * 16];

    int cnt = 0, firstN = 0;
    for (int base = 0; base < NPTS && cnt < 16; base += 32) {
        const int n = base + lane;
        const float* p = &xyz[(b * NPTS + n) * 3];
        const float dx = p[0] - qx, dy = p[1] - qy, dz = p[2] - qz;
        const float d2 = dx * dx + dy * dy + dz * dz;
        const unsigned mask = (unsigned)__ballot(d2 <= RAD2);
        if (cnt == 0 && mask) firstN = base + __ffs((int)mask) - 1;
        const bool in = (mask >> lane) & 1u;
        const int pos = cnt + __popc(mask & ((1u << lane) - 1u));
        if (in && pos < 16) row[pos] = n;
        cnt += __popc(mask);
    }
    const int c = cnt < 16 ? cnt : 16;
    if (lane < 16 && lane >= c) row[lane] = firstN;      // pad with first index
}

// ---------------------------------------------------------------------------
// Kernel 3: group (rel-xyz + gathered features) -> W1 GEMM (f32 WMMA) -> ReLU
// -> max over 16 neighbors. One 256-thread block per query point. Feature
// gather goes memory->LDS via async (per-lane scattered addresses); rel-xyz
// rows and the k=67 zero-pad row are computed on VALU. 8 waves x 17 k-slices
// of V_WMMA_F32_16X16X4_F32, then width-16 shfl_xor max over neighbors.
// ---------------------------------------------------------------------------
__global__ void __launch_bounds__(256)
group_w1_kernel(const float* __restrict__ xyz, const float* __restrict__ feat,
                const float* __restrict__ W1, const float* __restrict__ new_xyz,
                const int* __restrict__ nidx, float* __restrict__ sub_x)
{
    __shared__ float G[68 * 16];
    __shared__ float q[3];
    __shared__ int   nb[16];

    const int bs = blockIdx.x;                 // b*2048 + s
    const int b = bs >> 11, s = bs & 2047;
    const int tid = threadIdx.x;

    if (tid < 16) nb[tid] = nidx[bs * 16 + tid];
    if (tid < 3)  q[tid] = new_xyz[bs * 3 + tid];
    __syncthreads();

    // feature rows k=3..66: async global->LDS, per-lane gathered addresses
    for (int e = tid; e < 64 * 16; e += 256) {
        const int k = e >> 4, j = e & 15;
        async_g2l_b32(&feat[(b * 64 + k) * NPTS + nb[j]], &G[(k + 3) * 16 + j]);
    }
    // rel-xyz rows k=0..2 and zero-pad row k=67
    if (tid < 64) {
        const int k = tid >> 4, j = tid & 15;
        if (k < 3) G[k * 16 + j] = xyz[(b * NPTS + nb[j]) * 3 + k] - q[k];
        else       G[67 * 16 + j] = 0.0f;
    }
    async_wait0();
    __syncthreads();

    const int lane = tid & 31;
    const int m0 = (tid >> 5) * 16;            // wave m-tile
    const int r  = lane & 15;                  // A row / B col
    const int kg = lane >> 4;                  // k-group within slice

    v8f acc = {};
    for (int kk = 0; kk < 17; ++kk) {
        const int k0 = kk * 4 + kg * 2;
        v2f a, bf;
        a.x  = (k0     < 67) ? W1[(m0 + r) * 67 + k0]     : 0.0f;
        a.y  = (k0 + 1 < 67) ? W1[(m0 + r) * 67 + k0 + 1] : 0.0f;
        bf.x = G[k0 * 16 + r];
        bf.y = G[(k0 + 1) * 16 + r];
        acc = __builtin_amdgcn_wmma_f32_16x16x4_f32(
                  false, a, false, bf, (short)0, acc, false, false);
    }

#pragma unroll
    for (int j = 0; j < 8; ++j) {
        float v = acc[j] > 0.0f ? acc[j] : 0.0f;       // ReLU (commutes with max)
        for (int off = 8; off; off >>= 1) {
            const float o = __shfl_xor(v, off, 16);    // max over 16 neighbors
            v = v > o ? v : o;
        }
        if (r == 0)
            sub_x[(b * 128 + m0 + j + 8 * kg) * NPOINT + s] = v;
    }
}

// ---------------------------------------------------------------------------
// Kernel 4: 3-NN of each original point among the 2048 sampled points, with
// normalized inverse-distance weights. One thread per point.
// ---------------------------------------------------------------------------
__global__ void __launch_bounds__(256)
three_nn_kernel(const float* __restrict__ xyz, const float* __restrict__ new_xyz,
                int* __restrict__ idx3, float* __restrict__ w3)
{
    const int g = blockIdx.x * blockDim.x + threadIdx.x;    // b*N + n
    const int b = g >> 13;
    const float px = xyz[g * 3 + 0], py = xyz[g * 3 + 1], pz = xyz[g * 3 + 2];
    const float* q = &new_xyz[b * NPOINT * 3];

    float d0 = 3.4e38f, d1 = 3.4e38f, d2 = 3.4e38f;
    int   i0 = 0, i1 = 0, i2 = 0;
    for (int s = 0; s < NPOINT; ++s) {
        const float dx = q[s * 3] - px, dy = q[s * 3 + 1] - py, dz = q[s * 3 + 2] - pz;
        const float d = dx * dx + dy * dy + dz * dz;
        if (d < d0)      { d2 = d1; i2 = i1; d1 = d0; i1 = i0; d0 = d; i0 = s; }
        else if (d < d1) { d2 = d1; i2 = i1; d1 = d;  i1 = s; }
        else if (d < d2) { d2 = d;  i2 = s; }
    }
    float w0 = 1.0f / (d0 + 1e-8f), w1 = 1.0f / (d1 + 1e-8f), w2 = 1.0f / (d2 + 1e-8f);
    const float ws = w0 + w1 + w2;
    w0 /= ws; w1 /= ws; w2 /= ws;
    idx3[g * 3 + 0] = i0; idx3[g * 3 + 1] = i1; idx3[g * 3 + 2] = i2;
    w3[g * 3 + 0] = w0;  w3[g * 3 + 1] = w1;  w3[g * 3 + 2] = w2;
}

// ---------------------------------------------------------------------------
// Kernel 5: build cat = [features(64) ; interp(sub_x)(128)] tile [192x16] in
// LDS (feature rows streamed memory->LDS async while interp rows compute),
// then out = relu(W2[128x192] @ cat) via 8 waves x 48 f32-WMMA k-slices.
// One block per (b, 16-column tile).
// ---------------------------------------------------------------------------
__global__ void __launch_bounds__(256)
interp_w2_kernel(const float* __restrict__ feat, const float* __restrict__ W2,
                 const float* __restrict__ sub_x, const int* __restrict__ idx3,
                 const float* __restrict__ w3, float* __restrict__ out)
{
    __shared__ float C[192 * 16];

    const int bt = blockIdx.x;                  // b*512 + tile
    const int b = bt >> 9, n0 = (bt & 511) << 4;
    const int tid = threadIdx.x;

    for (int e = tid; e < 64 * 16; e += 256) {  // feature rows 0..63 (async DMA)
        const int k = e >> 4, j = e & 15;
        async_g2l_b32(&feat[(b * 64 + k) * NPTS + n0 + j], &C[e]);
    }
    for (int e = tid; e < 128 * 16; e += 256) { // interpolated rows 64..191
        const int c = e >> 4, j = e & 15;
        const int n = n0 + j;
        const int*   id = &idx3[(b * NPTS + n) * 3];
        const float* w  = &w3[(b * NPTS + n) * 3];
        const float* sx = &sub_x[(b * 128 + c) * NPOINT];
        C[(64 + c) * 16 + j] = w[0] * sx[id[0]] + w[1] * sx[id[1]] + w[2] * sx[id[2]];
    }
    async_wait0();
    __syncthreads();

    const int lane = tid & 31;
    const int m0 = (tid >> 5) * 16;
    const int r  = lane & 15;
    const int kg = lane >> 4;

    v8f acc = {};
    for (int kk = 0; kk < 48; ++kk) {           // K = 192 exactly
        const int k0 = kk * 4 + kg * 2;
        v2f a, bf;
        a.x  = W2[(m0 + r) * 192 + k0];
        a.y  = W2[(m0 + r) * 192 + k0 + 1];
        bf.x = C[k0 * 16 + r];
        bf.y = C[(k0 + 1) * 16 + r];
        acc = __builtin_amdgcn_wmma_f32_16x16x4_f32(
                  false, a, false, bf, (short)0, acc, false, false);
    }

#pragma unroll
    for (int j = 0; j < 8; ++j) {
        const float v = acc[j] > 0.0f ? acc[j] : 0.0f;
        out[(b * 128 + m0 + j + 8 * kg) * NPTS + n0 + r] = v;   // coalesced 16-wide
    }
}

// ---------------------------------------------------------------------------
extern "C" void kernel_launch(void* const* d_in, const int* in_sizes, int n_in,
                              void* d_out, int out_size, void* d_ws, size_t ws_size,
                              hipStream_t stream)
{
    (void)in_sizes; (void)n_in; (void)out_size; (void)ws_size;
    const float* xyz  = (const float*)d_in[0];   // [4,8192,3]
    const float* feat = (const float*)d_in[1];   // [4,64,8192]
    const float* W1   = (const float*)d_in[2];   // [128,67]
    const float* W2   = (const float*)d_in[3];   // [128,192]
    float* out = (float*)d_out;                  // [4,128,8192]

    char* ws = (char*)d_ws;                      // ~5.4 MB used
    float* new_xyz = (float*)(ws + 0);           // 4*2048*3 f32   (96 KB)
    int*   nidx    = (int*)  (ws + 98304);       // 4*2048*16 i32  (512 KB)
    float* sub_x   = (float*)(ws + 622592);      // 4*128*2048 f32 (4 MB)
    int*   idx3    = (int*)  (ws + 4816896);     // 4*8192*3 i32   (384 KB)
    float* w3      = (float*)(ws + 5210112);     // 4*8192*3 f32   (384 KB)

    fps_kernel      <<<BATCH, 1024, 0, stream>>>(xyz, new_xyz);
    ballquery_kernel<<<(BATCH * NPOINT) / 8, 256, 0, stream>>>(xyz, new_xyz, nidx);
    group_w1_kernel <<<BATCH * NPOINT, 256, 0, stream>>>(xyz, feat, W1, new_xyz, nidx, sub_x);
    three_nn_kernel <<<(BATCH * NPTS) / 256, 256, 0, stream>>>(xyz, new_xyz, idx3, w3);
    interp_w2_kernel<<<BATCH * (NPTS / 16), 256, 0, stream>>>(feat, W2, sub_x, idx3, w3, out);
}